// DWDN_16432544875235
// MI455X (gfx1250) — compile-verified
//
#include <hip/hip_runtime.h>
#include <math.h>

// Problem constants (from reference): B=4, NF=256, reduced C=64, H=W=256, KS=19
#define BB     4
#define NFEAT  256
#define CR     64
#define HH     256
#define WWID   256
#define KSZ    19
#define PADK   19
#define NP     294              // 256 + 2*19 (true fp/FFT domain)
#define LDP    320              // padded so 64x64 block tiles divide it (5*64)
#define PLANE  (LDP*LDP)        // 102400
#define NIMG   (BB*CR)          // 256 images of 294x294
#define NSPAT  (HH*WWID)        // 65536 spatial positions
#define KC     16               // k-chunk staged in LDS per block iteration

typedef __attribute__((ext_vector_type(2))) float v2f;
typedef __attribute__((ext_vector_type(8))) float v8f;

// D(16x16 f32) = A(16x4 f32) * B(4x16 f32) + C  — CDNA5 V_WMMA_F32_16X16X4_F32
__device__ __forceinline__ v8f wmma4(v2f a, v2f b, v8f c) {
  return __builtin_amdgcn_wmma_f32_16x16x4_f32(false, a, false, b, (short)0, c,
                                               false, false);
}

// ---------------------------------------------------------------------------
// DFT matrices: forward W[r,c] = e^{-2pi i rc/294}, inverse V[r,c] = e^{+..}/294
// Zeroed outside [0,294)^2 so the 320 padding is mathematically inert.
// ---------------------------------------------------------------------------
__global__ void k_dft_init(float* __restrict__ Wre, float* __restrict__ Wim,
                           float* __restrict__ Vre, float* __restrict__ Vim) {
  int idx = blockIdx.x * blockDim.x + threadIdx.x;
  if (idx >= PLANE) return;
  int r = idx / LDP, c = idx % LDP;
  float wr = 0.f, wi = 0.f, vr = 0.f, vi = 0.f;
  if (r < NP && c < NP) {
    int m = (int)(((long)r * (long)c) % NP);
    float t = -6.2831853071795864f * (float)m * (1.0f / (float)NP);
    float s, cs;
    __sincosf(t, &s, &cs);
    wr = cs; wi = s;
    vr = cs * (1.0f / (float)NP);
    vi = -s * (1.0f / (float)NP);
  }
  Wre[idx] = wr; Wim[idx] = wi; Vre[idx] = vr; Vim[idx] = vi;
}

// ---------------------------------------------------------------------------
// Channel reduce GEMM (WMMA): wf[b,o,p] = sum_c Wr[o,c] x[b,c,p] + br[o]
// written directly into interior of padded LDP-stride planes (img = b*64+o)
// ---------------------------------------------------------------------------
__global__ void k_reduce(const float* __restrict__ x, const float* __restrict__ Wr,
                         const float* __restrict__ br, float* __restrict__ planes) {
  int b = blockIdx.z;
  int rowBase = blockIdx.y << 4;   // o tile (M = 64)
  int colBase = blockIdx.x << 4;   // p tile (N = 65536)
  int lane = threadIdx.x;
  int mr = lane & 15, hi = lane >> 4;
  const float* xb = x + (long)b * NFEAT * NSPAT;
  v8f acc = {0.f,0.f,0.f,0.f,0.f,0.f,0.f,0.f};
  for (int k0 = 0; k0 < NFEAT; k0 += 4) {
    int ka = k0 + hi * 2;
    v2f a = *(const v2f*)(Wr + (rowBase + mr) * NFEAT + ka);
    v2f bv;
    bv.x = xb[(long)ka * NSPAT + colBase + mr];
    bv.y = xb[(long)(ka + 1) * NSPAT + colBase + mr];
    acc = wmma4(a, bv, acc);
  }
  int p = colBase + mr;
  int h = p >> 8, w = p & 255;
  long poff = (long)(PADK + h) * LDP + (PADK + w);
#pragma unroll
  for (int v = 0; v < 8; ++v) {
    int o = rowBase + v + 8 * hi;
    planes[(long)(b * CR + o) * PLANE + poff] = acc[v] + br[o];
  }
}

// ---------------------------------------------------------------------------
// Edge-replicate pad of the 294x294 fp domain; zero the LDP margin.
// ---------------------------------------------------------------------------
__global__ void k_pad(float* __restrict__ planes) {
  long gid = (long)blockIdx.x * blockDim.x + threadIdx.x;
  if (gid >= (long)NIMG * PLANE) return;
  int img = (int)(gid / PLANE);
  int idx = (int)(gid % PLANE);
  int y = idx / LDP, x = idx % LDP;
  if (y >= PADK && y < PADK + HH && x >= PADK && x < PADK + WWID) return; // interior
  float v = 0.f;
  if (y < NP && x < NP) {
    int sy = min(max(y, PADK), PADK + HH - 1);
    int sx = min(max(x, PADK), PADK + WWID - 1);
    v = planes[(long)img * PLANE + (long)sy * LDP + sx];
  }
  planes[(long)img * PLANE + idx] = v;
}

// ---------------------------------------------------------------------------
// OTF = fft2(psf2otf(kernel)): direct 19x19-tap DFT with circular shift folded
// into the phase: psf[(i-9) mod 294, (j-9) mod 294] = kernel[i,j]  (c-1 = 9)
// ---------------------------------------------------------------------------
__global__ void k_otf(const float* __restrict__ kern, float* __restrict__ otf_re,
                      float* __restrict__ otf_im) {
  int bb = blockIdx.y;
  int idx = blockIdx.x * blockDim.x + threadIdx.x;
  if (idx >= PLANE) return;
  int u = idx / LDP, v = idx % LDP;
  float sr = 0.f, si = 0.f;
  if (u < NP && v < NP) {
    const float* kb = kern + bb * (KSZ * KSZ);
    for (int i = 0; i < KSZ; ++i) {
      for (int j = 0; j < KSZ; ++j) {
        float kv = kb[i * KSZ + j];
        int m = (u * (i - 9) + v * (j - 9)) % NP;
        if (m < 0) m += NP;
        float t = -6.2831853071795864f * (float)m * (1.0f / (float)NP);
        float s, cs;
        __sincosf(t, &s, &cs);
        sr += kv * cs;
        si += kv * s;
      }
    }
  }
  otf_re[(long)bb * PLANE + idx] = sr;
  otf_im[(long)bb * PLANE + idx] = si;
}

// ---------------------------------------------------------------------------
// Wiener NSR: median3x3 (reflect) stats over the 294x294 fp domain
// ---------------------------------------------------------------------------
__device__ __forceinline__ void mnmx(float& a, float& b) {
  float t = fminf(a, b);
  b = fmaxf(a, b);
  a = t;
}
__device__ __forceinline__ float median9(float p0, float p1, float p2, float p3,
                                         float p4, float p5, float p6, float p7,
                                         float p8) {
  mnmx(p1, p2); mnmx(p4, p5); mnmx(p7, p8);
  mnmx(p0, p1); mnmx(p3, p4); mnmx(p6, p7);
  mnmx(p1, p2); mnmx(p4, p5); mnmx(p7, p8);
  mnmx(p0, p3); mnmx(p5, p8); mnmx(p4, p7);
  mnmx(p3, p6); mnmx(p1, p4); mnmx(p2, p5);
  mnmx(p4, p7); mnmx(p4, p2); mnmx(p6, p4);
  mnmx(p4, p2);
  return p4;
}
__device__ __forceinline__ int refl(int i) {
  return i < 0 ? -i : (i > NP - 1 ? 2 * (NP - 1) - i : i);
}

__global__ void k_nsr(const float* __restrict__ planes, float* __restrict__ nsr) {
  int img = blockIdx.x;
  const float* p = planes + (long)img * PLANE;
  int tid = threadIdx.x;
  float sd = 0.f, sd2 = 0.f, sx = 0.f, sx2 = 0.f;
  for (int i = tid; i < NP * NP; i += blockDim.x) {
    int y = i / NP, x = i % NP;
    int ym = refl(y - 1), yp = refl(y + 1);
    int xm = refl(x - 1), xp = refl(x + 1);
    float v = p[y * LDP + x];
    float med = median9(p[ym * LDP + xm], p[ym * LDP + x], p[ym * LDP + xp],
                        p[y  * LDP + xm], v,               p[y  * LDP + xp],
                        p[yp * LDP + xm], p[yp * LDP + x], p[yp * LDP + xp]);
    float d = med - v;
    sd += d; sd2 += d * d; sx += v; sx2 += v * v;
  }
  __shared__ float s0[256], s1[256], s2[256], s3[256];
  s0[tid] = sd; s1[tid] = sd2; s2[tid] = sx; s3[tid] = sx2;
  __syncthreads();
  for (int st = 128; st > 0; st >>= 1) {
    if (tid < st) {
      s0[tid] += s0[tid + st];
      s1[tid] += s1[tid + st];
      s2[tid] += s2[tid + st];
      s3[tid] += s3[tid + st];
    }
    __syncthreads();
  }
  if (tid == 0) {
    float n = (float)(NP * NP);
    float var_n = (s1[0] - s0[0] * s0[0] / n) / (n - 1.0f);
    float std_s = sqrtf((s3[0] - s2[0] * s2[0] / n) / (n - 1.0f));
    nsr[img] = var_n / std_s * (8.0f / 30.0f);
  }
}

// ---------------------------------------------------------------------------
// Generic complex GEMM over LDPxLDP padded planes (WMMA f32 16x16x4).
// 4-wave workgroup computes a 64x64 C macro-tile: per KC-chunk the block
// cooperatively stages a 64xKC A-slab and KCx64 B-slab in LDS (re+im), so each
// global fragment feeds 2 waves in each direction (halves L2 operand traffic);
// each wave then register-blocks a 32x32 tile (2x2 WMMA tiles) out of LDS.
// A-slab rows padded to 18 floats -> the 16-row-strided ds_load_b64 fragment
// reads land on distinct banks (18*mr mod 64 distinct for mr=0..15).
// C = A @ B (complex). HA/HB: imaginary part present. WI: produce C imag.
// stride == 0 means matrix shared across images (DFT matrices).
// f32 WMMA has no A/B NEG modifier (ISA NEG table) -> VALU-negate Aim for the
// Re accumulation.
// ---------------------------------------------------------------------------
template <bool HA, bool HB, bool WI>
__global__ void k_cgemm(const float* __restrict__ Are, const float* __restrict__ Aim,
                        const float* __restrict__ Bre, const float* __restrict__ Bim,
                        float* __restrict__ Cre, float* __restrict__ Cim,
                        long sA, long sB, long sC) {
  __shared__ float sAre[64][18];
  __shared__ float sAim[64][18];
  __shared__ float sBre[KC][64];
  __shared__ float sBim[KC][64];

  int img = blockIdx.z;
  int blockRow = blockIdx.y << 6;  // 64-row block tile
  int blockCol = blockIdx.x << 6;  // 64-col block tile
  int tid = threadIdx.x;           // 0..127
  int lane = tid & 31;
  int wid = tid >> 5;              // wave 0..3 in 2x2 grid
  int wr2 = (wid >> 1) << 5;       // wave row offset within block: 0 / 32
  int wc2 = (wid & 1) << 5;        // wave col offset within block: 0 / 32
  int mr = lane & 15, hi = lane >> 4;

  const float* are = Are + (long)img * sA;
  const float* aim = HA ? (Aim + (long)img * sA) : nullptr;
  const float* bre = Bre + (long)img * sB;
  const float* bim = HB ? (Bim + (long)img * sB) : nullptr;

  const v8f vzero = {0.f,0.f,0.f,0.f,0.f,0.f,0.f,0.f};
  v8f acc_re[2][2], acc_im[2][2];
#pragma unroll
  for (int r = 0; r < 2; ++r)
#pragma unroll
    for (int c = 0; c < 2; ++c) {
      acc_re[r][c] = vzero;
      acc_im[r][c] = vzero;
    }

  for (int kc = 0; kc < LDP; kc += KC) {
    // --- cooperative stage: A 64xKC, B KCx64 (re and, if present, im) ---
#pragma unroll
    for (int i = 0; i < (64 * KC) / 128; ++i) {  // 8 iters
      int idx = tid + i * 128;
      int row = idx >> 4;     // /KC  (KC==16)
      int kk  = idx & 15;
      long agoff = (long)(blockRow + row) * LDP + kc + kk;
      sAre[row][kk] = are[agoff];
      if (HA) sAim[row][kk] = aim[agoff];
      int krow = idx >> 6;    // /64
      int col  = idx & 63;
      long bgoff = (long)(kc + krow) * LDP + blockCol + col;
      sBre[krow][col] = bre[bgoff];
      if (HB) sBim[krow][col] = bim[bgoff];
    }
    __syncthreads();
    // --- compute KC/4 WMMA k-steps out of LDS ---
#pragma unroll
    for (int k4 = 0; k4 < KC; k4 += 4) {
      int ka = k4 + hi * 2;
      v2f aR[2], aI[2], naI[2], bR[2], bI[2];
#pragma unroll
      for (int r = 0; r < 2; ++r) {
        int arow = wr2 + 16 * r + mr;
        aR[r] = *(const v2f*)&sAre[arow][ka];
        if (HA) {
          aI[r] = *(const v2f*)&sAim[arow][ka];
          naI[r].x = -aI[r].x;
          naI[r].y = -aI[r].y;
        }
      }
#pragma unroll
      for (int c = 0; c < 2; ++c) {
        int col = wc2 + 16 * c + mr;
        bR[c].x = sBre[ka][col];
        bR[c].y = sBre[ka + 1][col];
        if (HB) {
          bI[c].x = sBim[ka][col];
          bI[c].y = sBim[ka + 1][col];
        }
      }
#pragma unroll
      for (int r = 0; r < 2; ++r) {
#pragma unroll
        for (int c = 0; c < 2; ++c) {
          acc_re[r][c] = wmma4(aR[r], bR[c], acc_re[r][c]);
          if (HA && HB)
            acc_re[r][c] = wmma4(naI[r], bI[c], acc_re[r][c]);  // Re -= Aim*Bim
          if (WI) {
            if (HB) acc_im[r][c] = wmma4(aR[r], bI[c], acc_im[r][c]);  // +Are*Bim
            if (HA) acc_im[r][c] = wmma4(aI[r], bR[c], acc_im[r][c]);  // +Aim*Bre
          }
        }
      }
    }
    __syncthreads();
  }

  int rowBase = blockRow + wr2;
  int colBase = blockCol + wc2;
  float* cre = Cre + (long)img * sC;
#pragma unroll
  for (int r = 0; r < 2; ++r)
#pragma unroll
    for (int c = 0; c < 2; ++c)
#pragma unroll
      for (int v = 0; v < 8; ++v)
        cre[(long)(rowBase + 16 * r + v + 8 * hi) * LDP + colBase + 16 * c + mr] =
            acc_re[r][c][v];
  if (WI) {
    float* cim = Cim + (long)img * sC;
#pragma unroll
    for (int r = 0; r < 2; ++r)
#pragma unroll
      for (int c = 0; c < 2; ++c)
#pragma unroll
        for (int v = 0; v < 8; ++v)
          cim[(long)(rowBase + 16 * r + v + 8 * hi) * LDP + colBase + 16 * c + mr] =
              acc_im[r][c][v];
  }
}

// ---------------------------------------------------------------------------
// Elementwise Wiener filter: G = conj(OTF)*F / (|OTF|^2 + nsr)
// ---------------------------------------------------------------------------
__global__ void k_filter(const float* __restrict__ Fre, const float* __restrict__ Fim,
                         const float* __restrict__ otf_re, const float* __restrict__ otf_im,
                         const float* __restrict__ nsr, float* __restrict__ Gre,
                         float* __restrict__ Gim) {
  long gid = (long)blockIdx.x * blockDim.x + threadIdx.x;
  if (gid >= (long)NIMG * PLANE) return;
  int img = (int)(gid / PLANE);
  int idx = (int)(gid % PLANE);
  int bb = img >> 6;  // img = b*64 + c
  float orr = otf_re[(long)bb * PLANE + idx];
  float oi  = otf_im[(long)bb * PLANE + idx];
  float fr = Fre[gid], fi = Fim[gid];
  float den = orr * orr + oi * oi + nsr[img];
  float inv = 1.0f / den;
  Gre[gid] = (orr * fr + oi * fi) * inv;
  Gim[gid] = (orr * fi - oi * fr) * inv;
}

// ---------------------------------------------------------------------------
// Channel expand GEMM (WMMA): out[b,o,p] = sum_c We[o,c]*clear[b,c,p] + be[o]
// clear read directly from the cropped region of the padded planes.
// ---------------------------------------------------------------------------
__global__ void k_expand(const float* __restrict__ planes, const float* __restrict__ We,
                         const float* __restrict__ be, float* __restrict__ out) {
  int b = blockIdx.z;
  int rowBase = blockIdx.y << 4;   // o tile (M = 256)
  int colBase = blockIdx.x << 4;   // p tile (N = 65536)
  int lane = threadIdx.x;
  int mr = lane & 15, hi = lane >> 4;
  int p = colBase + mr;
  int h = p >> 8, w = p & 255;
  long boff = (long)(PADK + h) * LDP + (PADK + w);
  v8f acc = {0.f,0.f,0.f,0.f,0.f,0.f,0.f,0.f};
  for (int k0 = 0; k0 < CR; k0 += 4) {
    int ka = k0 + hi * 2;
    v2f a = *(const v2f*)(We + (rowBase + mr) * CR + ka);
    v2f bv;
    bv.x = planes[(long)(b * CR + ka) * PLANE + boff];
    bv.y = planes[(long)(b * CR + ka + 1) * PLANE + boff];
    acc = wmma4(a, bv, acc);
  }
#pragma unroll
  for (int v = 0; v < 8; ++v) {
    int o = rowBase + v + 8 * hi;
    out[((long)b * NFEAT + o) * NSPAT + p] = acc[v] + be[o];
  }
}

// ---------------------------------------------------------------------------
extern "C" void kernel_launch(void* const* d_in, const int* in_sizes, int n_in,
                              void* d_out, int out_size, void* d_ws, size_t ws_size,
                              hipStream_t stream) {
  (void)in_sizes; (void)n_in; (void)out_size; (void)ws_size;
  const float* x        = (const float*)d_in[0];  // (4,256,256,256)
  const float* kern     = (const float*)d_in[1];  // (4,1,19,19)
  const float* w_reduce = (const float*)d_in[2];  // (64,256)
  const float* b_reduce = (const float*)d_in[3];  // (64,)
  const float* w_expand = (const float*)d_in[4];  // (256,64)
  const float* b_expand = (const float*)d_in[5];  // (256,)
  float* out = (float*)d_out;                      // (4,256,256,256)

  float* ws = (float*)d_ws;
  const size_t IP = (size_t)NIMG * PLANE;
  float* fp     = ws;                 // fp planes (re); reused for ifft real out
  float* Are    = fp    + IP;
  float* Aim    = Are   + IP;
  float* Bre_   = Aim   + IP;
  float* Bim_   = Bre_  + IP;
  float* Wre    = Bim_  + IP;         // forward DFT matrix
  float* Wim    = Wre   + PLANE;
  float* Vre    = Wim   + PLANE;      // inverse DFT matrix (includes 1/294)
  float* Vim    = Vre   + PLANE;
  float* otfre  = Vim   + PLANE;      // 4 * PLANE
  float* otfim  = otfre + (size_t)BB * PLANE;
  float* nsrv   = otfim + (size_t)BB * PLANE;  // 256 floats

  const dim3 gG(LDP / 64, LDP / 64, NIMG);  // 5 x 5 x 256 block macro-tiles

  // 0) DFT matrices
  k_dft_init<<<(PLANE + 255) / 256, 256, 0, stream>>>(Wre, Wim, Vre, Vim);
  // 1) channel reduce (WMMA) straight into padded planes
  k_reduce<<<dim3(NSPAT / 16, CR / 16, BB), 32, 0, stream>>>(x, w_reduce, b_reduce, fp);
  // 2) edge-replicate pad + zero margin
  k_pad<<<(unsigned)((IP + 255) / 256), 256, 0, stream>>>(fp);
  // 3) OTF of the blur kernels
  k_otf<<<dim3((PLANE + 255) / 256, BB), 256, 0, stream>>>(kern, otfre, otfim);
  // 4) Wiener NSR per (b,c)
  k_nsr<<<NIMG, 256, 0, stream>>>(fp, nsrv);
  // 5) forward FFT rows:  F1 = FP @ W       (real input)
  k_cgemm<false, true, true><<<gG, 128, 0, stream>>>(fp, nullptr, Wre, Wim,
                                                     Are, Aim, PLANE, 0, PLANE);
  // 6) forward FFT cols:  F2 = W @ F1
  k_cgemm<true, true, true><<<gG, 128, 0, stream>>>(Wre, Wim, Are, Aim,
                                                    Bre_, Bim_, 0, PLANE, PLANE);
  // 7) Wiener filter in frequency domain
  k_filter<<<(unsigned)((IP + 255) / 256), 256, 0, stream>>>(Bre_, Bim_, otfre, otfim,
                                                             nsrv, Are, Aim);
  // 8) inverse FFT cols:  C3 = V @ G
  k_cgemm<true, true, true><<<gG, 128, 0, stream>>>(Vre, Vim, Are, Aim,
                                                    Bre_, Bim_, 0, PLANE, PLANE);
  // 9) inverse FFT rows (real output only):  deblur = C3 @ V
  k_cgemm<true, true, false><<<gG, 128, 0, stream>>>(Bre_, Bim_, Vre, Vim,
                                                     fp, nullptr, PLANE, 0, PLANE);
  // 10) channel expand (WMMA) with crop + bias, into d_out
  k_expand<<<dim3(NSPAT / 16, NFEAT / 16, BB), 32, 0, stream>>>(fp, w_expand,
                                                                b_expand, out);
}